// PropagateAttention_3573412790319
// MI455X (gfx1250) — compile-verified
//
#include <hip/hip_runtime.h>
#include <hip/hip_bf16.h>

// Problem constants (from reference)
#define Bb 4
#define Nn 1024
#define Cc 768
#define Hh 12
#define DH 64
#define MROWS (Bb * Nn) // 4096
static constexpr float kScale = 0.125f; // 64^-0.5

typedef float v2f __attribute__((ext_vector_type(2)));
typedef float v8f __attribute__((ext_vector_type(8)));

// D(16x16 f32) = A(16x4 f32) * B(4x16 f32) + C  -- full fp32 WMMA
__device__ __forceinline__ v8f wmma4(v2f a, v2f b, v8f c) {
  return __builtin_amdgcn_wmma_f32_16x16x4_f32(false, a, false, b, (short)0, c,
                                               false, false);
}

// ---------------------------------------------------------------------------
// GEMM: Out[4096,768] = X[4096,768] @ W[768,768] (+ bias)
// 256 threads = 8 waves; wave w computes rows [blk.x*128 + w*16, +16),
// cols [blk.y*64, +64). W k-chunks staged to LDS transposed [col][k].
// ---------------------------------------------------------------------------
#define KCHUNK 32
#define LPAD 34

__launch_bounds__(256) __global__
void gemm768_wmma(const float* __restrict__ X, const float* __restrict__ W,
                  const float* __restrict__ bias, float* __restrict__ Out) {
  __shared__ float ldsW[64 * LPAD];
  const int tid = threadIdx.x;
  const int lane = tid & 31;
  const int wave = tid >> 5;
  const int mrow0 = blockIdx.x * 128 + wave * 16;
  const int nbase = blockIdx.y * 64;
  const int m = lane & 15;       // row (A) / col (B,C,D) within tile
  const int half = lane >> 4;    // k-half select
  const int kk0 = half * 2;

  v8f acc[4] = {};

  const int ldk = tid >> 3;        // 0..31 : k within chunk
  const int ldc = (tid & 7) * 8;   // 0..56 : col group

  for (int kc0 = 0; kc0 < Cc; kc0 += KCHUNK) {
    // cooperative load of W[kc0+ldk][nbase+ldc..+7] -> ldsW[col][k] (transposed)
    const float* wsrc = W + (size_t)(kc0 + ldk) * Cc + nbase + ldc;
    float4 w0 = *(const float4*)(wsrc);
    float4 w1 = *(const float4*)(wsrc + 4);
    if (kc0 + KCHUNK < Cc)
      __builtin_prefetch(wsrc + (size_t)KCHUNK * Cc, 0, 1);
    ldsW[(ldc + 0) * LPAD + ldk] = w0.x;
    ldsW[(ldc + 1) * LPAD + ldk] = w0.y;
    ldsW[(ldc + 2) * LPAD + ldk] = w0.z;
    ldsW[(ldc + 3) * LPAD + ldk] = w0.w;
    ldsW[(ldc + 4) * LPAD + ldk] = w1.x;
    ldsW[(ldc + 5) * LPAD + ldk] = w1.y;
    ldsW[(ldc + 6) * LPAD + ldk] = w1.z;
    ldsW[(ldc + 7) * LPAD + ldk] = w1.w;
    __syncthreads();

    const float* xrow = X + (size_t)(mrow0 + m) * Cc + kc0 + kk0;
#pragma unroll
    for (int k0 = 0; k0 < KCHUNK; k0 += 4) {
      v2f a = *(const v2f*)(xrow + k0);
#pragma unroll
      for (int t = 0; t < 4; ++t) {
        v2f b = *(const v2f*)(&ldsW[(t * 16 + m) * LPAD + k0 + kk0]);
        acc[t] = wmma4(a, b, acc[t]);
      }
    }
    __syncthreads();
  }

#pragma unroll
  for (int t = 0; t < 4; ++t) {
    const int col = nbase + t * 16 + m;
    const float bv = bias ? bias[col] : 0.0f;
#pragma unroll
    for (int j = 0; j < 8; ++j) {
      const int row = mrow0 + j + 8 * half;
      Out[(size_t)row * Cc + col] = acc[t][j] + bv;
    }
  }
}

// ---------------------------------------------------------------------------
// In-place L2 normalization of qs over each head's 64 channels.
// One wave per (row, head); lane handles 2 contiguous elements.
// ---------------------------------------------------------------------------
__launch_bounds__(256) __global__
void l2norm_heads(float* __restrict__ qs) {
  const int lane = threadIdx.x & 31;
  const int gw = (int)((blockIdx.x * blockDim.x + threadIdx.x) >> 5);
  const int row = gw / Hh;
  const int h = gw % Hh;
  float* p = qs + (size_t)row * Cc + h * DH + lane * 2;
  v2f v = *(v2f*)p;
  float ss = v.x * v.x + v.y * v.y;
#pragma unroll
  for (int off = 16; off >= 1; off >>= 1) ss += __shfl_xor(ss, off, 32);
  const float r = rsqrtf(ss);
  v.x *= r;
  v.y *= r;
  *(v2f*)p = v;
}

// ---------------------------------------------------------------------------
// Fused dual attention, single-pass online (flash) softmax.
// One block (8 waves) = one (b,h) and 8 consecutive 16-row query tiles.
// Per key tile nt: K / qsn(key) / Vc / Vs are cooperatively staged into LDS
// once per block (8x traffic reduction vs per-wave streaming):
//   ldsK/ldsS : row-major [16][KPADA]  -> B operand is one ds_load_b64
//   ldsVc/Vs  : transposed [64][VPAD]  -> B operand is one ds_load_b64
// S = qc.kc^T * 0.125 + gamma_h * (qsn.qsn^T); online softmax with per-tile
// row-max (16-lane shfl reduce) and exp-rescaled accumulators; P tile goes
// through per-wave LDS to convert C/D-layout -> A-layout for the P.V WMMAs.
// ---------------------------------------------------------------------------
#define KPADA 68
#define VPAD 18

__launch_bounds__(256) __global__
void attn_dual_wmma(const float* __restrict__ qc, const float* __restrict__ qsn,
                    const float* __restrict__ kc, const float* __restrict__ vc,
                    const float* __restrict__ vs,
                    const float* __restrict__ gamma, float* __restrict__ oc,
                    float* __restrict__ os) {
  __shared__ __align__(16) float ldsK[16 * KPADA];
  __shared__ __align__(16) float ldsS[16 * KPADA];
  __shared__ __align__(16) float ldsVc[64 * VPAD];
  __shared__ __align__(16) float ldsVs[64 * VPAD];
  __shared__ float ldsP[8 * 256];

  const int tid = threadIdx.x;
  const int lane = tid & 31;
  const int wave = tid >> 5;
  const int bh = blockIdx.x >> 3;           // 0..47
  const int mt = (blockIdx.x & 7) * 8 + wave; // query tile 0..63
  const int h = bh % Hh;
  const int b = bh / Hh;
  const int m = lane & 15;
  const int half = lane >> 4;
  const int kk0 = half * 2;
  const int qrow0 = b * Nn + mt * 16;
  const int hcol = h * DH;
  const float g = gamma[h];

  // staging indices (256 threads cover one 16x64 tile as float4s)
  const int skrow = tid >> 4;        // 0..15
  const int scol4 = (tid & 15) * 4;  // 0..60

  // Preload the 16x64 Q blocks (content + normalized semantic) as A operands.
  v2f aq[16], an[16];
  {
    const float* qp = qc + (size_t)(qrow0 + m) * Cc + hcol + kk0;
    const float* sp = qsn + (size_t)(qrow0 + m) * Cc + hcol + kk0;
#pragma unroll
    for (int s = 0; s < 16; ++s) {
      aq[s] = *(const v2f*)(qp + 4 * s);
      an[s] = *(const v2f*)(sp + 4 * s);
    }
  }

  float rmax[8], rsum[8];
#pragma unroll
  for (int j = 0; j < 8; ++j) {
    rmax[j] = -3.0e38f;
    rsum[j] = 0.0f;
  }
  v8f aoc[4] = {};
  v8f aos[4] = {};
  float* P = &ldsP[wave * 256];

  for (int nt = 0; nt < 64; ++nt) {
    // ---- cooperative staging of key-side tiles -------------------------
    {
      const size_t grow = (size_t)(b * Nn + nt * 16 + skrow) * Cc + hcol + scol4;
      float4 kv = *(const float4*)(kc + grow);
      float4 sv = *(const float4*)(qsn + grow);
      float4 vcv = *(const float4*)(vc + grow);
      float4 vsv = *(const float4*)(vs + grow);
      if (nt < 63) { // prefetch next key tile (global_prefetch_b8)
        __builtin_prefetch(kc + grow + (size_t)16 * Cc, 0, 1);
        __builtin_prefetch(vc + grow + (size_t)16 * Cc, 0, 1);
      }
      *(float4*)&ldsK[skrow * KPADA + scol4] = kv;
      *(float4*)&ldsS[skrow * KPADA + scol4] = sv;
      ldsVc[(scol4 + 0) * VPAD + skrow] = vcv.x;
      ldsVc[(scol4 + 1) * VPAD + skrow] = vcv.y;
      ldsVc[(scol4 + 2) * VPAD + skrow] = vcv.z;
      ldsVc[(scol4 + 3) * VPAD + skrow] = vcv.w;
      ldsVs[(scol4 + 0) * VPAD + skrow] = vsv.x;
      ldsVs[(scol4 + 1) * VPAD + skrow] = vsv.y;
      ldsVs[(scol4 + 2) * VPAD + skrow] = vsv.z;
      ldsVs[(scol4 + 3) * VPAD + skrow] = vsv.w;
    }
    __syncthreads();

    // ---- S tile: content + semantic WMMAs ------------------------------
    v8f ac = {};
    v8f av = {};
#pragma unroll
    for (int s = 0; s < 16; ++s) {
      v2f bk = *(const v2f*)(&ldsK[m * KPADA + 4 * s + kk0]);
      ac = wmma4(aq[s], bk, ac);
    }
#pragma unroll
    for (int s = 0; s < 16; ++s) {
      v2f bs = *(const v2f*)(&ldsS[m * KPADA + 4 * s + kk0]);
      av = wmma4(an[s], bs, av);
    }

    // ---- online softmax update ----------------------------------------
    float p[8], cf[8];
#pragma unroll
    for (int j = 0; j < 8; ++j) {
      const float sc = ac[j] * kScale + av[j] * g;
      float tm = sc;
#pragma unroll
      for (int off = 8; off >= 1; off >>= 1)
        tm = fmaxf(tm, __shfl_xor(tm, off, 16));
      const float nm = fmaxf(rmax[j], tm);
      cf[j] = __expf(rmax[j] - nm);
      rmax[j] = nm;
      p[j] = __expf(sc - nm);
      rsum[j] = rsum[j] * cf[j] + p[j];
      P[(j + 8 * half) * 16 + m] = p[j]; // [row][col] in LDS
    }
#pragma unroll
    for (int t = 0; t < 4; ++t)
#pragma unroll
      for (int j = 0; j < 8; ++j) {
        aoc[t][j] *= cf[j];
        aos[t][j] *= cf[j];
      }
    asm volatile("s_wait_dscnt 0x0" ::: "memory"); // P stores visible

    // ---- P.V accumulation ---------------------------------------------
#pragma unroll
    for (int s = 0; s < 4; ++s) {
      v2f pa = *(const v2f*)(P + m * 16 + 4 * s + kk0);
#pragma unroll
      for (int t = 0; t < 4; ++t) {
        v2f bc = *(const v2f*)(&ldsVc[(t * 16 + m) * VPAD + 4 * s + kk0]);
        aoc[t] = wmma4(pa, bc, aoc[t]);
        v2f bsv = *(const v2f*)(&ldsVs[(t * 16 + m) * VPAD + 4 * s + kk0]);
        aos[t] = wmma4(pa, bsv, aos[t]);
      }
    }
    asm volatile("s_wait_dscnt 0x0" ::: "memory"); // WAR on P before next iter
    __syncthreads(); // WAR on K/S/V tiles before restaging
  }

#pragma unroll
  for (int j = 0; j < 8; ++j) {
#pragma unroll
    for (int off = 8; off >= 1; off >>= 1)
      rsum[j] += __shfl_xor(rsum[j], off, 16);
  }

#pragma unroll
  for (int t = 0; t < 4; ++t) {
    const int col = hcol + t * 16 + m;
#pragma unroll
    for (int j = 0; j < 8; ++j) {
      const int row = qrow0 + j + 8 * half;
      const float inv = 1.0f / rsum[j];
      oc[(size_t)row * Cc + col] = aoc[t][j] * inv;
      os[(size_t)row * Cc + col] = aos[t][j] * inv;
    }
  }
}

// ---------------------------------------------------------------------------
extern "C" void kernel_launch(void* const* d_in, const int* in_sizes, int n_in,
                              void* d_out, int out_size, void* d_ws,
                              size_t ws_size, hipStream_t stream) {
  (void)in_sizes; (void)n_in; (void)out_size; (void)ws_size;
  const float* fc = (const float*)d_in[0];
  const float* fs = (const float*)d_in[1];
  const float* Wqc = (const float*)d_in[2];
  const float* Wqs = (const float*)d_in[3];
  const float* Wkc = (const float*)d_in[4];
  const float* Wvc = (const float*)d_in[5];
  const float* Wvs = (const float*)d_in[6];
  const float* gamma = (const float*)d_in[7];
  const float* Wpc = (const float*)d_in[8];
  const float* bpc = (const float*)d_in[9];
  const float* Wps = (const float*)d_in[10];
  const float* bps = (const float*)d_in[11];

  float* out = (float*)d_out;
  float* ws = (float*)d_ws;
  const size_t S = (size_t)MROWS * Cc; // 3,145,728 floats per buffer

  float* qcb = ws + 0 * S;
  float* qsb = ws + 1 * S; // normalized in place
  float* kcb = ws + 2 * S;
  float* vcb = ws + 3 * S;
  float* vsb = ws + 4 * S;
  float* ocb = ws + 5 * S;
  float* osb = ws + 6 * S;

  const dim3 gg(MROWS / 128, Cc / 64); // 32 x 12
  const dim3 bb(256);

  hipLaunchKernelGGL(gemm768_wmma, gg, bb, 0, stream, fc, Wqc, (const float*)nullptr, qcb);
  hipLaunchKernelGGL(gemm768_wmma, gg, bb, 0, stream, fs, Wqs, (const float*)nullptr, qsb);
  hipLaunchKernelGGL(gemm768_wmma, gg, bb, 0, stream, fc, Wkc, (const float*)nullptr, kcb);
  hipLaunchKernelGGL(gemm768_wmma, gg, bb, 0, stream, fc, Wvc, (const float*)nullptr, vcb);
  hipLaunchKernelGGL(gemm768_wmma, gg, bb, 0, stream, fs, Wvs, (const float*)nullptr, vsb);

  hipLaunchKernelGGL(l2norm_heads, dim3((MROWS * Hh) / 8), bb, 0, stream, qsb);

  // 48 (b,h) pairs x 8 query-tile groups = 384 blocks of 8 waves
  hipLaunchKernelGGL(attn_dual_wmma, dim3(Bb * Hh * 8), bb, 0, stream, qcb, qsb,
                     kcb, vcb, vsb, gamma, ocb, osb);

  hipLaunchKernelGGL(gemm768_wmma, gg, bb, 0, stream, ocb, Wpc, bpc, out);
  hipLaunchKernelGGL(gemm768_wmma, gg, bb, 0, stream, osb, Wps, bps, out + S);
}